// weighted_graph_conv_5927054868789
// MI455X (gfx1250) — compile-verified
//
#include <hip/hip_runtime.h>
#include <hip/hip_bf16.h>

// Problem constants (from the reference)
#define GN   20000      // nodes
#define GT   4          // time steps
#define GF   64         // features in/out
#define GE   640000     // edges
#define ROWS (GN * GT)  // 80000 rows for the linear layer

typedef __attribute__((ext_vector_type(2))) float v2f;
typedef __attribute__((ext_vector_type(8))) float v8f;

// ---------------------------------------------------------------------------
// Kernel 1: zero the h accumulator in workspace (float4 stores)
// ---------------------------------------------------------------------------
__global__ __launch_bounds__(256) void wgc_zero(float4* __restrict__ p, int n4) {
    int i = blockIdx.x * 256 + threadIdx.x;
    if (i < n4) p[i] = make_float4(0.f, 0.f, 0.f, 0.f);
}

// ---------------------------------------------------------------------------
// Kernel 2: edge gather + weighted scatter-add into h (L2-resident atomics).
// One thread handles one (edge, t, 4-feature quad): 1x float4 load, 4 fadds.
// Thread layout: tid = e*64 + t*16 + quad  (T*F/4 = 64 work items per edge)
// ---------------------------------------------------------------------------
__global__ __launch_bounds__(256) void wgc_edge_scatter(
    const float*  __restrict__ nf,   // (N,T,F)
    const float*  __restrict__ ew,   // (T,E)
    const int*    __restrict__ src,  // (E,)
    const int*    __restrict__ dst,  // (E,)
    float*        __restrict__ h)    // (N,T,F) accumulator
{
    int tid = blockIdx.x * 256 + threadIdx.x;
    int e = tid >> 6;
    if (e >= GE) return;
    int r = tid & 63;
    int t = r >> 4;
    int f = (r & 15) << 2;

    int s = src[e];
    int d = dst[e];
    float w = ew[t * GE + e];

    const float4 x = *reinterpret_cast<const float4*>(nf + s * (GT * GF) + t * GF + f);
    float* hp = h + d * (GT * GF) + t * GF + f;

    // no-return fp32 atomic adds -> GLOBAL_ATOMIC_ADD_F32 (RMW at L2)
    unsafeAtomicAdd(hp + 0, x.x * w);
    unsafeAtomicAdd(hp + 1, x.y * w);
    unsafeAtomicAdd(hp + 2, x.z * w);
    unsafeAtomicAdd(hp + 3, x.w * w);
}

// ---------------------------------------------------------------------------
// Kernel 3: out = h @ W^T + b via V_WMMA_F32_16X16X4_F32 (exact fp32).
// One wave32 computes a 16-row x 64-col strip: 4 accumulators, 16 K-steps.
// W (64x64 = 16KB) is staged in LDS by the block.
//
// A fragment (16x4 fp32, 2 VGPRs/lane):
//   lanes 0-15 : row M = m0+lane, K = {k0, k0+1}
//   lanes 16-31: row M = m0+lane-16, K = {k0+2, k0+3}
// B fragment (4x16 fp32): B[k][n] = W[n][k], mirrored lane layout.
// D (16x16 fp32, 8 VGPRs): vgpr r -> M = r (lanes 0-15), M = r+8 (lanes 16-31),
//   N = lane & 15.
// ---------------------------------------------------------------------------
__global__ __launch_bounds__(128) void wgc_gemm_wmma(
    const float* __restrict__ h,     // (ROWS, 64)
    const float* __restrict__ Wm,    // (64, 64) row-major, W[n][k]
    const float* __restrict__ bias,  // (64,)
    float*       __restrict__ out,   // (ROWS, 64)
    int strips)
{
    __shared__ float sW[GF * GF];    // 16 KB

    // cooperative load of W into LDS: 4096 floats / 128 threads = 8 x float4
    for (int i = threadIdx.x; i < (GF * GF) / 4; i += 128) {
        reinterpret_cast<float4*>(sW)[i] = reinterpret_cast<const float4*>(Wm)[i];
    }
    __syncthreads();

    const int wave = threadIdx.x >> 5;
    const int lane = threadIdx.x & 31;
    const int strip = blockIdx.x * 4 + wave;
    if (strip >= strips) return;              // wave-uniform exit (EXEC all-1s for WMMA)

    const int m0   = strip * 16;
    const int half = lane >> 4;               // 0: K pair {0,1}, 1: K pair {2,3}
    const int lid  = lane & 15;

    v8f acc0 = {}, acc1 = {}, acc2 = {}, acc3 = {};

    const float* arow = h + (m0 + lid) * GF + half * 2;

#pragma unroll
    for (int k0 = 0; k0 < GF; k0 += 4) {
        v2f a = *reinterpret_cast<const v2f*>(arow + k0);

        v2f b0 = *reinterpret_cast<const v2f*>(sW + (0 * 16 + lid) * GF + k0 + half * 2);
        v2f b1 = *reinterpret_cast<const v2f*>(sW + (1 * 16 + lid) * GF + k0 + half * 2);
        v2f b2 = *reinterpret_cast<const v2f*>(sW + (2 * 16 + lid) * GF + k0 + half * 2);
        v2f b3 = *reinterpret_cast<const v2f*>(sW + (3 * 16 + lid) * GF + k0 + half * 2);

        acc0 = __builtin_amdgcn_wmma_f32_16x16x4_f32(false, a, false, b0, (short)0, acc0, false, false);
        acc1 = __builtin_amdgcn_wmma_f32_16x16x4_f32(false, a, false, b1, (short)0, acc1, false, false);
        acc2 = __builtin_amdgcn_wmma_f32_16x16x4_f32(false, a, false, b2, (short)0, acc2, false, false);
        acc3 = __builtin_amdgcn_wmma_f32_16x16x4_f32(false, a, false, b3, (short)0, acc3, false, false);
    }

    // bias + store per the D layout
    const float bv0 = bias[0 * 16 + lid];
    const float bv1 = bias[1 * 16 + lid];
    const float bv2 = bias[2 * 16 + lid];
    const float bv3 = bias[3 * 16 + lid];

#pragma unroll
    for (int r = 0; r < 8; ++r) {
        const int m = m0 + half * 8 + r;
        float* orow = out + m * GF;
        orow[0 * 16 + lid] = acc0[r] + bv0;
        orow[1 * 16 + lid] = acc1[r] + bv1;
        orow[2 * 16 + lid] = acc2[r] + bv2;
        orow[3 * 16 + lid] = acc3[r] + bv3;
    }
}

// ---------------------------------------------------------------------------
extern "C" void kernel_launch(void* const* d_in, const int* in_sizes, int n_in,
                              void* d_out, int out_size, void* d_ws, size_t ws_size,
                              hipStream_t stream) {
    const float* nf   = (const float*)d_in[0];   // (N,T,F_IN)
    const float* ew   = (const float*)d_in[1];   // (T,E)
    const int*   src  = (const int*)d_in[2];     // (E,)
    const int*   dst  = (const int*)d_in[3];     // (E,)
    const float* Wm   = (const float*)d_in[4];   // (F_OUT,F_IN)
    const float* bias = (const float*)d_in[5];   // (F_OUT,)
    float* out = (float*)d_out;                  // (N,T,F_OUT)
    float* h   = (float*)d_ws;                   // (N,T,F_IN) accumulator, 20.5 MB

    // 1) zero h
    {
        const int n4 = ROWS * GF / 4;            // 1,280,000 float4
        const int blocks = (n4 + 255) / 256;
        wgc_zero<<<blocks, 256, 0, stream>>>((float4*)h, n4);
    }

    // 2) edge scatter-add
    {
        const long long total = (long long)GE * 64;   // 40,960,000 threads
        const int blocks = (int)((total + 255) / 256);
        wgc_edge_scatter<<<blocks, 256, 0, stream>>>(nf, ew, src, dst, h);
    }

    // 3) linear layer with fp32 WMMA
    {
        const int strips = ROWS / 16;            // 5000
        const int blocks = (strips + 3) / 4;     // 4 waves per block
        wgc_gemm_wmma<<<blocks, 128, 0, stream>>>(h, Wm, bias, out, strips);
    }
}